// ParticleNet_90108413870688
// MI455X (gfx1250) — compile-verified
//
#include <hip/hip_runtime.h>
#include <hip/hip_bf16.h>
#include <math.h>

// ---------------------------------------------------------------------------
// ParticleNet-style block on MI455X (gfx1250, wave32).
// fp32 end-to-end; GEMMs via V_WMMA_F32_16X16X4_F32; KNN replaced by exact
// dense ball-query (out-of-ball weights are exactly zero per the reference).
// ---------------------------------------------------------------------------

typedef __attribute__((ext_vector_type(2))) float v2f;
typedef __attribute__((ext_vector_type(8))) float v8f;

#define BATCH  4
#define NPTS   4096
#define IN_C   64
#define OUT_C  128
#define MID_C  32
#define DIR_C  (MID_C * 6)   // 192

#define R2F        0.0016f          // RADIUS^2
#define DR2F       0.0064f          // DECAY_RADIUS^2
#define INV_DRR    (1.0f / (DR2F - R2F))
#define EPS_BN     1e-5f

// ---------------------------------------------------------------------------
// GEMM: C[M x N] = A[M x K] * W[K x N] + bias[N] (+ addC[M x N] if non-null)
// One wave per 16x16 output tile. fp32 WMMA, K stepped by 4.
// ---------------------------------------------------------------------------
__global__ void gemm_f32_wmma(const float* __restrict__ A,
                              const float* __restrict__ W,
                              const float* __restrict__ bias,
                              const float* __restrict__ addC,
                              float* __restrict__ Cout,
                              int M, int N, int Kdim)
{
    const int gtid  = blockIdx.x * blockDim.x + threadIdx.x;
    const int wave  = gtid >> 5;
    const int lane  = threadIdx.x & 31;
    const int half  = lane >> 4;       // 0: lanes 0-15, 1: lanes 16-31
    const int l16   = lane & 15;

    const int tilesN = N >> 4;
    const int tm = wave / tilesN;
    const int tn = wave - tm * tilesN;
    if (tm * 16 >= M) return;          // uniform per wave -> EXEC stays all-1s

    v8f acc = {0.f, 0.f, 0.f, 0.f, 0.f, 0.f, 0.f, 0.f};

    // A fragment row for this lane (both halves cover M = l16, different K).
    const float* Arow = A + (size_t)(tm * 16 + l16) * Kdim;
    const int    ncol = tn * 16 + l16;

    for (int k0 = 0; k0 < Kdim; k0 += 4) {
        const int ka = k0 + 2 * half;
        v2f afrag;                      // A: VGPR0=K ka, VGPR1=K ka+1 (contiguous)
        afrag.x = Arow[ka];
        afrag.y = Arow[ka + 1];
        v2f bfrag;                      // B: rows ka / ka+1 at column ncol
        bfrag.x = W[(size_t)ka * N + ncol];
        bfrag.y = W[(size_t)(ka + 1) * N + ncol];
        acc = __builtin_amdgcn_wmma_f32_16x16x4_f32(
            /*neg_a=*/false, afrag, /*neg_b=*/false, bfrag,
            /*c_mod=*/(short)0, acc, /*reuse_a=*/false, /*reuse_b=*/false);
    }

    const float bv = bias ? bias[ncol] : 0.f;
#pragma unroll
    for (int r = 0; r < 8; ++r) {
        const int m = tm * 16 + r + 8 * half;
        float v = acc[r] + bv;
        if (addC) v += addC[(size_t)m * N + ncol];
        Cout[(size_t)m * N + ncol] = v;
    }
}

// ---------------------------------------------------------------------------
// Dense ball-query aggregation. One wave per query point; lane d accumulates
// channel d (MID_C == 32 == wave32). xyz for the batch staged in LDS (48 KB).
// Grid: (NPTS/32, BATCH), block 256 (8 waves, 4 queries each).
// ---------------------------------------------------------------------------
__global__ void aggregate_kernel(const float* __restrict__ xyz,
                                 const float* __restrict__ direc6,
                                 float* __restrict__ agg)
{
    __shared__ float xs[NPTS], ys[NPTS], zs[NPTS];

    const int b = blockIdx.y;
    const float* xb = xyz + (size_t)b * NPTS * 3;
    for (int j = threadIdx.x; j < NPTS; j += blockDim.x) {
        xs[j] = xb[3 * j + 0];
        ys[j] = xb[3 * j + 1];
        zs[j] = xb[3 * j + 2];
    }
    __syncthreads();

    const int lane = threadIdx.x & 31;
    const int wv   = threadIdx.x >> 5;           // 0..7
    const float* dbase = direc6 + (size_t)b * NPTS * DIR_C;

    for (int t = 0; t < 4; ++t) {
        const int i = blockIdx.x * 32 + wv * 4 + t;
        const float qx = xs[i], qy = ys[i], qz = zs[i];
        float acc = 0.f;                          // channel `lane`
        float S   = 0.f;                          // replicated dw sum

        for (int base = 0; base < NPTS; base += 32) {
            const int j = base + lane;
            const float dx = xs[j] - qx;
            const float dy = ys[j] - qy;
            const float dz = zs[j] - qz;
            const float r2 = dx * dx + dy * dy + dz * dz;
            unsigned mask = (unsigned)__ballot(r2 < DR2F);
            while (mask) {
                const int tt = __builtin_ctz(mask);
                mask &= mask - 1;
                const float bdx = __shfl(dx, tt);
                const float bdy = __shfl(dy, tt);
                const float bdz = __shfl(dz, tt);
                const float br2 = __shfl(r2, tt);
                const float dw  = 1.f - (br2 - R2F) * INV_DRR;   // > 0 in ball
                S += dw;
                const float inv = 1.f / (sqrtf(br2) + 1e-8f);
                // axes: +z,-z,+y,-y,+x,-x ; cos = max(dot,0)/(|v|+eps), squared
                const float c0 = fmaxf(bdz, 0.f) * inv;
                const float c1 = fmaxf(-bdz, 0.f) * inv;
                const float c2 = fmaxf(bdy, 0.f) * inv;
                const float c3 = fmaxf(-bdy, 0.f) * inv;
                const float c4 = fmaxf(bdx, 0.f) * inv;
                const float c5 = fmaxf(-bdx, 0.f) * inv;
                const float* dr = dbase + (size_t)(base + tt) * DIR_C + lane * 6;
                const float s6 = dr[0] * c0 * c0 + dr[1] * c1 * c1 +
                                 dr[2] * c2 * c2 + dr[3] * c3 * c3 +
                                 dr[4] * c4 * c4 + dr[5] * c5 * c5;
                acc += s6 * dw;
            }
        }
        agg[((size_t)b * NPTS + i) * MID_C + lane] = acc / S;
    }
}

// ---------------------------------------------------------------------------
// BN statistics: one block (256 thr) per channel, deterministic tree reduce.
// ---------------------------------------------------------------------------
__global__ void bn_stats_kernel(const float* __restrict__ X,
                                float* __restrict__ mean,
                                float* __restrict__ var,
                                int R, int C)
{
    __shared__ float ssum[256];
    __shared__ float ssq[256];
    const int c = blockIdx.x;
    float s = 0.f, q = 0.f;
    for (int r = threadIdx.x; r < R; r += 256) {
        const float v = X[(size_t)r * C + c];
        s += v;
        q += v * v;
    }
    ssum[threadIdx.x] = s;
    ssq[threadIdx.x]  = q;
    __syncthreads();
    for (int st = 128; st > 0; st >>= 1) {
        if (threadIdx.x < st) {
            ssum[threadIdx.x] += ssum[threadIdx.x + st];
            ssq[threadIdx.x]  += ssq[threadIdx.x + st];
        }
        __syncthreads();
    }
    if (threadIdx.x == 0) {
        const float mu = ssum[0] / (float)R;
        mean[c] = mu;
        var[c]  = ssq[0] / (float)R - mu * mu;   // biased var (ddof=0)
    }
}

// ---------------------------------------------------------------------------
// y = relu((x - mu) / sqrt(var + eps) * g + beta)
// ---------------------------------------------------------------------------
__global__ void bn_relu_kernel(const float* __restrict__ X,
                               const float* __restrict__ mean,
                               const float* __restrict__ var,
                               const float* __restrict__ g,
                               const float* __restrict__ beta,
                               float* __restrict__ Y,
                               int total, int C)
{
    const int idx = blockIdx.x * blockDim.x + threadIdx.x;
    if (idx >= total) return;
    const int c = idx % C;
    const float v = (X[idx] - mean[c]) * rsqrtf(var[c] + EPS_BN) * g[c] + beta[c];
    Y[idx] = fmaxf(v, 0.f);
}

// ---------------------------------------------------------------------------
extern "C" void kernel_launch(void* const* d_in, const int* in_sizes, int n_in,
                              void* d_out, int out_size, void* d_ws, size_t ws_size,
                              hipStream_t stream)
{
    const float* xyz    = (const float*)d_in[0];
    const float* points = (const float*)d_in[1];
    const float* Wc     = (const float*)d_in[2];
    const float* bc     = (const float*)d_in[3];
    const float* Wd     = (const float*)d_in[4];
    const float* bd     = (const float*)d_in[5];
    const float* g1     = (const float*)d_in[6];
    const float* beta1  = (const float*)d_in[7];
    const float* W2     = (const float*)d_in[8];
    const float* b2     = (const float*)d_in[9];
    const float* g2     = (const float*)d_in[10];
    const float* beta2  = (const float*)d_in[11];
    float* out = (float*)d_out;

    const int M = BATCH * NPTS;                 // 16384 rows

    // workspace layout (floats)
    float* ws     = (float*)d_ws;
    float* center = ws;                          // M * OUT_C
    float* direc6 = center + (size_t)M * OUT_C;  // M * DIR_C
    float* agg    = direc6 + (size_t)M * DIR_C;  // M * MID_C  (BN'd in place)
    float* z      = agg    + (size_t)M * MID_C;  // M * OUT_C
    float* mean1  = z      + (size_t)M * OUT_C;  // MID_C
    float* var1   = mean1 + MID_C;               // MID_C
    float* mean2  = var1  + MID_C;               // OUT_C
    float* var2   = mean2 + OUT_C;               // OUT_C
    (void)ws_size; (void)in_sizes; (void)n_in; (void)out_size;

    // 1) center = points @ Wc + bc     (tiles: 1024 x 8 -> 8192 waves)
    {
        const int waves = (M / 16) * (OUT_C / 16);
        gemm_f32_wmma<<<waves / 8, 256, 0, stream>>>(points, Wc, bc, nullptr,
                                                     center, M, OUT_C, IN_C);
    }
    // 2) direc6 = points @ Wd + bd    (tiles: 1024 x 12 -> 12288 waves)
    {
        const int waves = (M / 16) * (DIR_C / 16);
        gemm_f32_wmma<<<waves / 8, 256, 0, stream>>>(points, Wd, bd, nullptr,
                                                     direc6, M, DIR_C, IN_C);
    }
    // 3) ball-query aggregation -> agg (M x 32)
    {
        dim3 grid(NPTS / 32, BATCH);
        aggregate_kernel<<<grid, 256, 0, stream>>>(xyz, direc6, agg);
    }
    // 4) BN1 stats + 5) BN1 apply + relu (in place)
    bn_stats_kernel<<<MID_C, 256, 0, stream>>>(agg, mean1, var1, M, MID_C);
    bn_relu_kernel<<<(M * MID_C) / 256, 256, 0, stream>>>(agg, mean1, var1,
                                                          g1, beta1, agg,
                                                          M * MID_C, MID_C);
    // 6) z = agg @ W2 + b2 + center
    {
        const int waves = (M / 16) * (OUT_C / 16);
        gemm_f32_wmma<<<waves / 8, 256, 0, stream>>>(agg, W2, b2, center,
                                                     z, M, OUT_C, MID_C);
    }
    // 7) BN2 stats + 8) BN2 apply + relu -> d_out
    bn_stats_kernel<<<OUT_C, 256, 0, stream>>>(z, mean2, var2, M, OUT_C);
    bn_relu_kernel<<<(M * OUT_C) / 256, 256, 0, stream>>>(z, mean2, var2,
                                                          g2, beta2, out,
                                                          M * OUT_C, OUT_C);
}